// torso_left_right_actor_17781164605718
// MI455X (gfx1250) — compile-verified
//
#include <hip/hip_runtime.h>
#include <math.h>

// ---------------- problem constants ----------------
#define N_NODES   100000
#define N_EDGES   1600000
#define N_GRAPHS  512
#define IN_DIM    11
#define HID       64
#define OUT_DIM   16
#define ROWTILES  (N_NODES / 16)            // 6250
#define SOFTPLUS_BIAS 0.5413248546129181f   // log(expm1(1.0))
#define SCALE_LB  1e-4f

typedef float v2f __attribute__((ext_vector_type(2)));
typedef float v8f __attribute__((ext_vector_type(8)));

// ---------------- zero fill ----------------
__global__ __launch_bounds__(256) void k_zero(float* __restrict__ p, int n) {
    int t = blockIdx.x * blockDim.x + threadIdx.x;
    if (t < n) p[t] = 0.0f;
}

// ---------------- input linear: h = x @ W0 + b0 ----------------
__global__ __launch_bounds__(256) void k_lin0(const float* __restrict__ x,
                                              const float* __restrict__ W0,
                                              const float* __restrict__ b0,
                                              float* __restrict__ h) {
    int t = blockIdx.x * blockDim.x + threadIdx.x;
    if (t >= N_NODES * HID) return;
    int node = t >> 6;
    int col  = t & 63;
    const float* xr = x + node * IN_DIM;
    float acc = b0[col];
#pragma unroll
    for (int i = 0; i < IN_DIM; ++i)
        acc = fmaf(xr[i], W0[i * HID + col], acc);
    h[t] = acc;
}

// ---------------- edge scatter-add: agg[dst] += h[src] ----------------
// thread handles one edge x 4 feature dims (float4 gather, 4 f32 atomics).
// agg (25.6MB) lives in L2 (192MB) so atomics resolve in-cache.
__global__ __launch_bounds__(256) void k_scatter(const float* __restrict__ h,
                                                 const int* __restrict__ src,
                                                 const int* __restrict__ dst,
                                                 float* __restrict__ agg) {
    int t = blockIdx.x * blockDim.x + threadIdx.x;
    if (t >= N_EDGES * (HID / 4)) return;
    int e = t >> 4;             // HID/4 == 16 slices per edge
    int d = (t & 15) << 2;
    int s  = src[e];
    int dd = dst[e];
    const float4 v = *(const float4*)(h + (size_t)s * HID + d);
    float* base = agg + (size_t)dd * HID + d;
    atomicAdd(base + 0, v.x);
    atomicAdd(base + 1, v.y);
    atomicAdd(base + 2, v.z);
    atomicAdd(base + 3, v.w);
}

// ---------------- fp32 WMMA GEMM: out = tanh(A1@W1 [+ A2@W2] + bias) -----
// A1/A2: [N_NODES, 64] fp32, W: [64, NCOLS]. One wave owns a 16-row slab and
// ALL col tiles (A read from global exactly once per slab). Weights staged in
// LDS by the whole block, B fragments served by ds_load -> no VMEM wait in
// front of each V_WMMA_F32_16X16X4_F32. K=64 in 16 steps of 4, fully unrolled.
template<int NCOLS, bool HAS2>
__global__ __launch_bounds__(256) void k_gemm_wmma(const float* __restrict__ A1,
                                                   const float* __restrict__ W1,
                                                   const float* __restrict__ A2,
                                                   const float* __restrict__ W2,
                                                   const float* __restrict__ bias,
                                                   float* __restrict__ out) {
    constexpr int CT = NCOLS / 16;              // col tiles per wave
    __shared__ float lw1[HID * NCOLS];
    __shared__ float lw2[HAS2 ? HID * NCOLS : 1];

    // cooperative weight stage (block-wide), then barrier
    for (int i = threadIdx.x; i < HID * NCOLS; i += 256) {
        lw1[i] = W1[i];
        if (HAS2) lw2[i] = W2[i];
    }
    __syncthreads();

    const int waveInBlock = threadIdx.x >> 5;
    const int rowTile = blockIdx.x * 8 + waveInBlock;
    if (rowTile >= ROWTILES) return;            // wave-uniform, after barrier
    const int lane = threadIdx.x & 31;

    const int m    = lane & 15;                 // A row / B col within tile
    const int koff = (lane >> 4) << 1;          // lanes 16-31 hold K+2,K+3

    v8f acc[CT];
#pragma unroll
    for (int ct = 0; ct < CT; ++ct) acc[ct] = (v8f){0.f,0.f,0.f,0.f,0.f,0.f,0.f,0.f};

    const float* a1row = A1 + (size_t)(rowTile * 16 + m) * HID;
    const float* a2row = HAS2 ? (A2 + (size_t)(rowTile * 16 + m) * HID) : nullptr;

#pragma unroll
    for (int k0 = 0; k0 < HID; k0 += 4) {
        const int kb = k0 + koff;
        v2f a1; a1.x = a1row[kb];               // A: M=m, K=kb / kb+1 (b64)
                a1.y = a1row[kb + 1];
#pragma unroll
        for (int ct = 0; ct < CT; ++ct) {
            v2f b; b.x = lw1[kb * NCOLS + ct * 16 + m];      // B from LDS
                   b.y = lw1[(kb + 1) * NCOLS + ct * 16 + m];
            acc[ct] = __builtin_amdgcn_wmma_f32_16x16x4_f32(false, a1, false, b,
                                                            (short)0, acc[ct],
                                                            false, false);
        }
        if (HAS2) {
            v2f a2; a2.x = a2row[kb];
                    a2.y = a2row[kb + 1];
#pragma unroll
            for (int ct = 0; ct < CT; ++ct) {
                v2f b; b.x = lw2[kb * NCOLS + ct * 16 + m];
                       b.y = lw2[(kb + 1) * NCOLS + ct * 16 + m];
                acc[ct] = __builtin_amdgcn_wmma_f32_16x16x4_f32(false, a2, false, b,
                                                                (short)0, acc[ct],
                                                                false, false);
            }
        }
    }

    // C/D layout: VGPR v holds row (mbase+v), N = lane&15 (lanes 16-31: M+8)
    const int n     = lane & 15;
    const int mbase = (lane >> 4) << 3;
#pragma unroll
    for (int ct = 0; ct < CT; ++ct) {
        const float bb = bias[ct * 16 + n];
#pragma unroll
        for (int v = 0; v < 8; ++v) {
            float val = acc[ct][v] + bb;
            out[(size_t)(rowTile * 16 + mbase + v) * NCOLS + ct * 16 + n] = tanhf(val);
        }
    }
}

// ---------------- pooling: segment sums + counts over batch ----------------
__global__ __launch_bounds__(256) void k_pool(const float* __restrict__ h3,
                                              const int* __restrict__ batch,
                                              float* __restrict__ sums,
                                              float* __restrict__ cnts) {
    int t = blockIdx.x * blockDim.x + threadIdx.x;
    if (t >= N_NODES * OUT_DIM) return;
    int node = t >> 4;
    int d    = t & 15;
    int b    = batch[node];
    atomicAdd(&sums[b * OUT_DIM + d], h3[t]);
    if (d == 0) atomicAdd(&cnts[b], 1.0f);
}

// ---------------- head: mean, split, biased softplus ----------------
__global__ __launch_bounds__(256) void k_final(const float* __restrict__ sums,
                                               const float* __restrict__ cnts,
                                               float* __restrict__ out) {
    int t = blockIdx.x * blockDim.x + threadIdx.x;
    if (t >= N_GRAPHS * 8) return;
    int g = t >> 3;
    int j = t & 7;
    float cnt = fmaxf(cnts[g], 1.0f);
    float loc = sums[g * OUT_DIM + j] / cnt;
    float sr  = sums[g * OUT_DIM + 8 + j] / cnt;
    out[t] = loc;                               // loc: out[0 .. 4095]
    float xx = sr + SOFTPLUS_BIAS;
    float sp = log1pf(expf(-fabsf(xx))) + fmaxf(xx, 0.0f);   // stable softplus
    out[N_GRAPHS * 8 + t] = fmaxf(sp, SCALE_LB); // scale: out[4096 .. 8191]
}

static inline int cdiv(int a, int b) { return (a + b - 1) / b; }

extern "C" void kernel_launch(void* const* d_in, const int* in_sizes, int n_in,
                              void* d_out, int out_size, void* d_ws, size_t ws_size,
                              hipStream_t stream) {
    // setup_inputs() order
    const float* x      = (const float*)d_in[0];
    const int*   eidx   = (const int*)d_in[1];       // [2, N_EDGES] (int32 on wire)
    const int*   batch  = (const int*)d_in[2];
    const float* W0     = (const float*)d_in[3];
    const float* b0     = (const float*)d_in[4];
    const float* Wrel1  = (const float*)d_in[5];
    const float* brel1  = (const float*)d_in[6];
    const float* Wroot1 = (const float*)d_in[7];
    const float* Wrel2  = (const float*)d_in[8];
    const float* brel2  = (const float*)d_in[9];
    const float* Wroot2 = (const float*)d_in[10];
    const float* W3     = (const float*)d_in[11];
    const float* b3     = (const float*)d_in[12];
    float* out = (float*)d_out;

    const int* src = eidx;
    const int* dst = eidx + N_EDGES;

    // workspace carve-up
    char* ws = (char*)d_ws;
    const size_t FEAT_BYTES = (size_t)N_NODES * HID * sizeof(float);   // 25.6 MB
    float* h0   = (float*)(ws);                                        // h / reused for h2
    float* agg  = (float*)(ws + FEAT_BYTES);
    float* h1   = (float*)(ws + 2 * FEAT_BYTES);
    float* h3   = (float*)(ws + 3 * FEAT_BYTES);                       // [N,16]
    float* sums = (float*)(ws + 3 * FEAT_BYTES + (size_t)N_NODES * OUT_DIM * sizeof(float));
    float* cnts = sums + N_GRAPHS * OUT_DIM;

    const int B = 256;
    const int featThreads = N_NODES * HID;                 // 6.4M
    const int edgeThreads = N_EDGES * (HID / 4);           // 25.6M
    const int gemmBlocks  = cdiv(ROWTILES, 8);             // 8 waves/block, 1 slab/wave

    // h0 = x@W0 + b0
    k_lin0<<<cdiv(featThreads, B), B, 0, stream>>>(x, W0, b0, h0);

    // ---- GraphConv layer 1 ----
    k_zero<<<cdiv(featThreads, B), B, 0, stream>>>(agg, featThreads);
    k_scatter<<<cdiv(edgeThreads, B), B, 0, stream>>>(h0, src, dst, agg);
    k_gemm_wmma<HID, true><<<gemmBlocks, B, 0, stream>>>(agg, Wrel1, h0, Wroot1,
                                                         brel1, h1);

    // ---- GraphConv layer 2 ----
    k_zero<<<cdiv(featThreads, B), B, 0, stream>>>(agg, featThreads);
    k_scatter<<<cdiv(edgeThreads, B), B, 0, stream>>>(h1, src, dst, agg);
    k_gemm_wmma<HID, true><<<gemmBlocks, B, 0, stream>>>(agg, Wrel2, h1, Wroot2,
                                                         brel2, h0);        // -> h0 (reuse)

    // ---- output linear: h3 = tanh(h0@W3 + b3) ----
    k_gemm_wmma<OUT_DIM, false><<<gemmBlocks, B, 0, stream>>>(h0, W3, nullptr, nullptr,
                                                              b3, h3);

    // ---- mean pool + head ----
    k_zero<<<cdiv(N_GRAPHS * OUT_DIM + N_GRAPHS, B), B, 0, stream>>>(
        sums, N_GRAPHS * OUT_DIM + N_GRAPHS);
    k_pool<<<cdiv(N_NODES * OUT_DIM, B), B, 0, stream>>>(h3, batch, sums, cnts);
    k_final<<<cdiv(N_GRAPHS * 8, B), B, 0, stream>>>(sums, cnts, out);
}